// SuperPointMatchingMutualTopk_2173253452299
// MI455X (gfx1250) — compile-verified
//
#include <hip/hip_runtime.h>
#include <hip/hip_bf16.h>
#include <math.h>

typedef __attribute__((ext_vector_type(2))) float v2f;
typedef __attribute__((ext_vector_type(8))) float v8f;

#define D_DIM   256
#define KSTEPS  (D_DIM / 4)      // 64 wmma f32 16x16x4 steps per tile
#define TOPK    3
#define ROWS_PER_BLOCK 64        // 4 waves x 16 rows
#define BLOCK_THREADS  128
#define TILE_N  16
#define BSTRIDE 260              // LDS row stride (floats): conflict-free frag reads
#define EPS_F   1e-8f

// ---------------------------------------------------------------------------
// Fused GEMM (A @ B^T over D=256) + per-row running top-3 of the dot product.
// B tiles are DMA'd into LDS with gfx1250 GLOBAL_LOAD_ASYNC_TO_LDS_B128
// (ASYNCcnt-tracked), double-buffered under the WMMA compute.
// ---------------------------------------------------------------------------
__global__ __launch_bounds__(BLOCK_THREADS, 1)
void fused_gemm_rowtop3(const float* __restrict__ A, const float* __restrict__ B,
                        float* __restrict__ topv, int* __restrict__ topi,
                        int ncols)
{
    __shared__ float sB[2][TILE_N * BSTRIDE];                   // 33280 B
    __shared__ float mergeV[ROWS_PER_BLOCK * 16 * TOPK];        // 12288 B
    __shared__ int   mergeI[ROWS_PER_BLOCK * 16 * TOPK];        // 12288 B

    const int tid  = threadIdx.x;
    const int wave = tid >> 5;
    const int lane = tid & 31;
    const int lrow = lane & 15;     // M for A-frags, N for B-frags/C
    const int half = lane >> 4;     // selects K pair (A/B) and M half (C)

    const int blockRow = blockIdx.x * ROWS_PER_BLOCK;
    const int myRow    = blockRow + wave * 16 + lrow;

    // ---- preload this wave's whole A panel (16 rows x 256) into registers ----
    // ISA 16x4 f32 A layout: lanes 0-15 hold K=0,1 (VGPR0,1); lanes 16-31 K=2,3
    v2f afrag[KSTEPS];
    {
        const float* ap = A + (size_t)myRow * D_DIM + half * 2;
        #pragma unroll
        for (int kk = 0; kk < KSTEPS; ++kk)
            afrag[kk] = *(const v2f*)(ap + kk * 4);
    }

    // per-lane running top-3 dots for the 8 C rows this lane holds
    float tv[8][TOPK];
    int   ti[8][TOPK];
    #pragma unroll
    for (int v = 0; v < 8; ++v)
        #pragma unroll
        for (int k = 0; k < TOPK; ++k) { tv[v][k] = -2.0f; ti[v][k] = -1; }

    const int numTiles = ncols / TILE_N;

    // per-thread fixed (row, col4) pattern: 1024 float4 per tile / 128 thr = 8
    // async DMA: global -> LDS, no VGPR staging, tracked by ASYNCcnt
    auto loadB_async = [&](int ct, int buf) {
        #pragma unroll
        for (int i = 0; i < (TILE_N * D_DIM / 4) / BLOCK_THREADS; ++i) {
            int e   = tid + i * BLOCK_THREADS;   // float4 index in tile
            int row = e >> 6;                    // 64 float4 per row
            int c4  = e & 63;
            unsigned ldsb = (unsigned)(size_t)(&sB[buf][row * BSTRIDE + c4 * 4]);
            unsigned goff = (unsigned)(((ct * TILE_N + row) * D_DIM + c4 * 4) *
                                       (int)sizeof(float));
            asm volatile("global_load_async_to_lds_b128 %0, %1, %2"
                         :
                         : "v"(ldsb), "v"(goff), "s"(B)
                         : "memory");
        }
    };

    loadB_async(0, 0);
    asm volatile("s_wait_asynccnt 0x0" ::: "memory");
    __syncthreads();

    for (int ct = 0; ct < numTiles; ++ct) {
        const int cur = ct & 1;
        if (ct + 1 < numTiles) loadB_async(ct + 1, cur ^ 1);
        if (ct + 2 < numTiles) {
            // gfx1250 global_prefetch_b8: pull the tile-after-next toward GL2
            const float* pf = B + (size_t)(ct + 2) * TILE_N * D_DIM + tid * 32;
            __builtin_prefetch(pf, 0, 0);
        }

        // B 4x16 f32 frag: lane -> N, (half,vgpr) -> K, mirrored from A layout
        const float* bp = &sB[cur][lrow * BSTRIDE + half * 2];
        v8f c = {};
        #pragma unroll
        for (int kk = 0; kk < KSTEPS; ++kk) {
            v2f b = *(const v2f*)(bp + kk * 4);
            c = __builtin_amdgcn_wmma_f32_16x16x4_f32(
                    false, afrag[kk], false, b, (short)0, c, false, false);
        }

        // C layout: VGPR v, lanes 0-15 -> M=v, N=lane; lanes 16-31 -> M=v+8
        const int col = ct * TILE_N + lrow;
        #pragma unroll
        for (int v = 0; v < 8; ++v) {
            float val = c[v];
            if (val > tv[v][2]) {
                if (val > tv[v][0]) {
                    tv[v][2]=tv[v][1]; ti[v][2]=ti[v][1];
                    tv[v][1]=tv[v][0]; ti[v][1]=ti[v][0];
                    tv[v][0]=val;      ti[v][0]=col;
                } else if (val > tv[v][1]) {
                    tv[v][2]=tv[v][1]; ti[v][2]=ti[v][1];
                    tv[v][1]=val;      ti[v][1]=col;
                } else {
                    tv[v][2]=val;      ti[v][2]=col;
                }
            }
        }

        // DMA for the next tile must be complete (and all waves past their
        // reads of the buffer it wrote) before the next iteration uses it.
        asm volatile("s_wait_asynccnt 0x0" ::: "memory");
        __syncthreads();
    }

    // ---- merge the 16 per-lane candidate sets of each row ----
    #pragma unroll
    for (int v = 0; v < 8; ++v) {
        int localRow = wave * 16 + half * 8 + v;
        #pragma unroll
        for (int k = 0; k < TOPK; ++k) {
            mergeV[(localRow * 16 + lrow) * TOPK + k] = tv[v][k];
            mergeI[(localRow * 16 + lrow) * TOPK + k] = ti[v][k];
        }
    }
    __syncthreads();

    if (tid < ROWS_PER_BLOCK) {
        float b0=-2.f, b1=-2.f, b2=-2.f;
        int   j0=-1,   j1=-1,   j2=-1;
        for (int l = 0; l < 16; ++l) {
            #pragma unroll
            for (int k = 0; k < TOPK; ++k) {
                float val = mergeV[(tid * 16 + l) * TOPK + k];
                int   idx = mergeI[(tid * 16 + l) * TOPK + k];
                if (val > b2) {
                    if (val > b0)      { b2=b1;j2=j1; b1=b0;j1=j0; b0=val;j0=idx; }
                    else if (val > b1) { b2=b1;j2=j1; b1=val;j1=idx; }
                    else               { b2=val;j2=idx; }
                }
            }
        }
        int g = blockRow + tid;
        topv[g*TOPK+0]=b0; topv[g*TOPK+1]=b1; topv[g*TOPK+2]=b2;
        topi[g*TOPK+0]=j0; topi[g*TOPK+1]=j1; topi[g*TOPK+2]=j2;
    }
}

// ---------------------------------------------------------------------------
// Mutual check: keep row-candidate c iff r is among c's column top-3.
// Matched cols sorted ascending (jnp.nonzero scans columns in order).
// ---------------------------------------------------------------------------
__global__ __launch_bounds__(256)
void match_mutual(const float* __restrict__ topv_r, const int* __restrict__ topi_r,
                  const int* __restrict__ topi_c,
                  int* __restrict__ mcol, float* __restrict__ mscore,
                  int* __restrict__ mcount, int n)
{
    int r = blockIdx.x * blockDim.x + threadIdx.x;
    if (r >= n) return;
    int   cols[TOPK];
    float sc[TOPK];
    int cnt = 0;
    #pragma unroll
    for (int k = 0; k < TOPK; ++k) {
        int   c   = topi_r[r*TOPK+k];
        float dot = topv_r[r*TOPK+k];
        bool mut = (topi_c[c*TOPK+0] == r) | (topi_c[c*TOPK+1] == r) |
                   (topi_c[c*TOPK+2] == r);
        if (mut) {
            cols[cnt] = c;
            sc[cnt]   = sqrtf(2.0f - 2.0f * dot + EPS_F);
            ++cnt;
        }
    }
    for (int a = 0; a < cnt; ++a)
        for (int b = a + 1; b < cnt; ++b)
            if (cols[b] < cols[a]) {
                int   tc = cols[a]; cols[a] = cols[b]; cols[b] = tc;
                float ts = sc[a];   sc[a]   = sc[b];   sc[b]   = ts;
            }
    mcount[r] = cnt;
    for (int k = 0; k < cnt; ++k) {
        mcol[r*TOPK+k]   = cols[k];
        mscore[r*TOPK+k] = sc[k];
    }
}

// ---------------------------------------------------------------------------
// Exclusive prefix sum of per-row match counts (single block).
// ---------------------------------------------------------------------------
__global__ __launch_bounds__(256)
void scan_counts(const int* __restrict__ mcount, int* __restrict__ moff, int n)
{
    __shared__ int part[256];
    int t = threadIdx.x;
    int per = (n + 255) / 256;
    int s = 0;
    for (int i = 0; i < per; ++i) {
        int r = t * per + i;
        if (r < n) s += mcount[r];
    }
    part[t] = s;
    __syncthreads();
    if (t == 0) {
        int run = 0;
        for (int i = 0; i < 256; ++i) { int x = part[i]; part[i] = run; run += x; }
    }
    __syncthreads();
    int run = part[t];
    for (int i = 0; i < per; ++i) {
        int r = t * per + i;
        if (r < n) { moff[r] = run; run += mcount[r]; }
    }
}

__global__ __launch_bounds__(256)
void init_out(float* __restrict__ orow, float* __restrict__ ocol,
              float* __restrict__ osc, int total)
{
    int i = blockIdx.x * blockDim.x + threadIdx.x;
    if (i < total) { orow[i] = -1.0f; ocol[i] = -1.0f; osc[i] = 0.0f; }
}

__global__ __launch_bounds__(256)
void scatter_out(const int* __restrict__ mcol, const float* __restrict__ mscore,
                 const int* __restrict__ mcount, const int* __restrict__ moff,
                 float* __restrict__ orow, float* __restrict__ ocol,
                 float* __restrict__ osc, int n)
{
    int r = blockIdx.x * blockDim.x + threadIdx.x;
    if (r >= n) return;
    int cnt = mcount[r];
    int off = moff[r];
    for (int k = 0; k < cnt; ++k) {
        orow[off+k] = (float)r;
        ocol[off+k] = (float)mcol[r*TOPK+k];
        osc[off+k]  = mscore[r*TOPK+k];
    }
}

extern "C" void kernel_launch(void* const* d_in, const int* in_sizes, int n_in,
                              void* d_out, int out_size, void* d_ws, size_t ws_size,
                              hipStream_t stream)
{
    const float* src = (const float*)d_in[0];   // (n, 256)
    const float* tgt = (const float*)d_in[1];   // (m, 256)
    const int n = in_sizes[0] / D_DIM;
    const int m = in_sizes[1] / D_DIM;
    const int maxCorr = n * TOPK;

    // workspace carve-out
    char* ws = (char*)d_ws;
    float* topv_r = (float*)ws;  ws += (size_t)n * TOPK * sizeof(float);
    int*   topi_r = (int*)ws;    ws += (size_t)n * TOPK * sizeof(int);
    float* topv_c = (float*)ws;  ws += (size_t)m * TOPK * sizeof(float);
    int*   topi_c = (int*)ws;    ws += (size_t)m * TOPK * sizeof(int);
    int*   mcol   = (int*)ws;    ws += (size_t)n * TOPK * sizeof(int);
    float* mscore = (float*)ws;  ws += (size_t)n * TOPK * sizeof(float);
    int*   mcount = (int*)ws;    ws += (size_t)n * sizeof(int);
    int*   moff   = (int*)ws;    ws += (size_t)n * sizeof(int);

    float* orow = (float*)d_out;          // rows as floats
    float* ocol = orow + maxCorr;         // cols as floats
    float* osc  = ocol + maxCorr;         // scores

    // row top-3 of S = src.tgt^T, then col top-3 via the transposed product
    fused_gemm_rowtop3<<<n / ROWS_PER_BLOCK, BLOCK_THREADS, 0, stream>>>(
        src, tgt, topv_r, topi_r, m);
    fused_gemm_rowtop3<<<m / ROWS_PER_BLOCK, BLOCK_THREADS, 0, stream>>>(
        tgt, src, topv_c, topi_c, n);

    match_mutual<<<(n + 255) / 256, 256, 0, stream>>>(
        topv_r, topi_r, topi_c, mcol, mscore, mcount, n);
    scan_counts<<<1, 256, 0, stream>>>(mcount, moff, n);
    init_out<<<(maxCorr + 255) / 256, 256, 0, stream>>>(orow, ocol, osc, maxCorr);
    scatter_out<<<(n + 255) / 256, 256, 0, stream>>>(
        mcol, mscore, mcount, moff, orow, ocol, osc, n);
}